// RelativeExtensionFusion_22393959481733
// MI455X (gfx1250) — compile-verified
//
#include <hip/hip_runtime.h>
#include <hip/hip_bf16.h>
#include <math.h>

typedef __attribute__((ext_vector_type(2))) float v2f;
typedef __attribute__((ext_vector_type(4))) float v4f;
typedef __attribute__((ext_vector_type(8))) float v8f;

#define B_  64
#define N_  256
#define DV_ 2048
#define DP_ 8
#define DT_ 768
#define DF_ 512
#define DO_ (DV_ + DP_)   // 2056

__device__ __forceinline__ float gelu_exact(float x) {
    // jax.nn.gelu(approximate=False): 0.5*x*(1+erf(x/sqrt(2)))
    return 0.5f * x * (1.0f + erff(x * 0.7071067811865475f));
}

// ---------------------------------------------------------------------------
// Kernel 1: query GEMM  (M=64 batch, K=768, N=512) with V_WMMA_F32_16X16X4_F32
// writes qw[b,f] = gelu(text0 @ Wt + bt) * Wa[f]
// ---------------------------------------------------------------------------
__global__ __launch_bounds__(32)
void k_query_wmma(const float* __restrict__ text, const float* __restrict__ Wt,
                  const float* __restrict__ bt,   const float* __restrict__ Wa,
                  float* __restrict__ qw) {
    const int m0   = blockIdx.x * 16;   // batch tile
    const int n0   = blockIdx.y * 16;   // DF tile
    const int lane = threadIdx.x;       // exactly one wave32 -> EXEC all ones
    const int half = lane >> 4;
    const int lmod = lane & 15;

    const float* arow = text + (size_t)(m0 + lmod) * (N_ * DT_);  // text[b,0,:]
    v8f acc = {};
    for (int k0 = 0; k0 < DT_; k0 += 4) {
        const int ka = k0 + 2 * half;
        v2f a = *(const v2f*)(arow + ka);                 // A[m][ka], A[m][ka+1]
        v2f b;
        b.x = Wt[(size_t)(ka + 0) * DF_ + (n0 + lmod)];   // B[ka][n]
        b.y = Wt[(size_t)(ka + 1) * DF_ + (n0 + lmod)];
        acc = __builtin_amdgcn_wmma_f32_16x16x4_f32(
            false, a, false, b, (short)0, acc, false, false);
    }
    const int   f   = n0 + lmod;
    const float btv = bt[f];
    const float wav = Wa[f];
#pragma unroll
    for (int r = 0; r < 8; ++r) {
        const int bb = m0 + r + 8 * half;                 // C/D: M = r + 8*(l/16)
        const float x = acc[r] + btv;
        qw[(size_t)bb * DF_ + f] = gelu_exact(x) * wav;
    }
}

// ---------------------------------------------------------------------------
// Kernel 2: V GEMM  (M=2056 rows of Wo, K=512, N=64 batches) with fp32 WMMA
// V[b, c] = sum_f Wo[c,f] * qw[b,f]
// ---------------------------------------------------------------------------
__global__ __launch_bounds__(32)
void k_vmat_wmma(const float* __restrict__ Wo, const float* __restrict__ qw,
                 float* __restrict__ V) {
    const int m0   = blockIdx.x * 16;   // c tile (last tile partial: 2056..2063)
    const int n0   = blockIdx.y * 16;   // batch tile
    const int lane = threadIdx.x;
    const int half = lane >> 4;
    const int lmod = lane & 15;

    const int  row   = m0 + lmod;
    const bool rowok = (row < DO_);
    const float mask = rowok ? 1.0f : 0.0f;               // v_cndmask, no EXEC games
    const float* arow = Wo + (size_t)(rowok ? row : 0) * DF_;
    const float* brow = qw + (size_t)(n0 + lmod) * DF_;

    v8f acc = {};
    for (int k0 = 0; k0 < DF_; k0 += 4) {
        const int ka = k0 + 2 * half;
        v2f a = *(const v2f*)(arow + ka);
        a.x *= mask; a.y *= mask;
        v2f b = *(const v2f*)(brow + ka);                 // qw is [b][f]: contiguous in k
        acc = __builtin_amdgcn_wmma_f32_16x16x4_f32(
            false, a, false, b, (short)0, acc, false, false);
    }
#pragma unroll
    for (int r = 0; r < 8; ++r) {
        const int c  = m0 + r + 8 * half;
        const int bb = n0 + lmod;
        if (c < DO_) V[(size_t)bb * DO_ + c] = acc[r];
    }
}

// ---------------------------------------------------------------------------
// Small kernels: column sums of position, and per-batch scalar c[b]
// ---------------------------------------------------------------------------
__global__ void k_possum(const float* __restrict__ pos, float* __restrict__ S) {
    const int t = blockIdx.x * blockDim.x + threadIdx.x;  // 0..511
    if (t >= B_ * DP_) return;
    const int b = t >> 3, k = t & 7;
    const float* p = pos + (size_t)b * N_ * DP_ + k;
    float s = 0.f;
    for (int n = 0; n < N_; ++n) s += p[n * DP_];
    S[t] = s;
}

__global__ void k_cb(const float* __restrict__ qw, const float* __restrict__ bo,
                     const float* __restrict__ ba, float* __restrict__ cb) {
    const int b = blockIdx.x * blockDim.x + threadIdx.x;
    if (b >= B_) return;
    const float* q = qw + (size_t)b * DF_;
    float s = 0.f;
    for (int f = 0; f < DF_; ++f) s += bo[f] * q[f];
    cb[b] = s + ba[0];
}

// ---------------------------------------------------------------------------
// Kernel 3 (the HBM-bound one): one wave32 per output row (b,n)
// out[b,n] = visual[b,n,:] . V[b,0:2048]  (V staged in LDS, reused by 8 rows)
//          + sum_k (257*pos[b,n,k] - S[b,k]) * V[b,2048+k]  + cb[b]
// visual is 134 MB streamed exactly once -> non-temporal loads so the small
// reused V/L2-resident operands are not evicted.
// ---------------------------------------------------------------------------
__global__ __launch_bounds__(256)
void k_final(const float* __restrict__ visual, const float* __restrict__ pos,
             const float* __restrict__ V, const float* __restrict__ S,
             const float* __restrict__ cb, float* __restrict__ out) {
    __shared__ float sV[DV_];                 // 8 KB of the WGP's 320 KB LDS
    const int b   = blockIdx.x >> 5;          // 32 blocks per batch
    const int ng  = blockIdx.x & 31;
    const int tid = threadIdx.x;

    const float* Vb = V + (size_t)b * DO_;
#pragma unroll
    for (int i = 0; i < DV_ / 256; ++i)       // cooperative stage: 8 floats/thread
        sV[i * 256 + tid] = Vb[i * 256 + tid];
    __syncthreads();

    const int wave = tid >> 5;
    const int lane = tid & 31;
    const int n    = ng * 8 + wave;

    const float* vr = visual + (size_t)(b * N_ + n) * DV_;
    float acc = 0.f;
#pragma unroll
    for (int i = 0; i < DV_ / (32 * 4); ++i) {        // 16x b128 loads per lane
        const int idx = (i * 32 + lane) * 4;
        const v4f x = __builtin_nontemporal_load((const v4f*)(vr + idx)); // NT b128
        const v4f w = *(const v4f*)(&sV[idx]);                            // ds_load_b128
        acc += x.x * w.x + x.y * w.y + x.z * w.z + x.w * w.w;
    }
    if (lane < DP_) {                          // fold rel-position part into lanes 0..7
        const float p = pos[(size_t)(b * N_ + n) * DP_ + lane];
        acc += (257.0f * p - S[b * DP_ + lane]) * Vb[DV_ + lane];
    }
    if (lane == 0) acc += cb[b];
#pragma unroll
    for (int off = 16; off > 0; off >>= 1)     // wave32 reduction
        acc += __shfl_down(acc, off, 32);
    if (lane == 0) out[(size_t)b * N_ + n] = acc;
}

// ---------------------------------------------------------------------------
extern "C" void kernel_launch(void* const* d_in, const int* in_sizes, int n_in,
                              void* d_out, int out_size, void* d_ws, size_t ws_size,
                              hipStream_t stream) {
    const float* visual   = (const float*)d_in[0];
    const float* position = (const float*)d_in[1];
    const float* text     = (const float*)d_in[2];
    const float* Wt       = (const float*)d_in[3];
    const float* bt       = (const float*)d_in[4];
    const float* Wo       = (const float*)d_in[5];
    const float* bo       = (const float*)d_in[6];
    const float* Wa       = (const float*)d_in[7];
    const float* ba       = (const float*)d_in[8];
    float* out = (float*)d_out;

    char*  ws = (char*)d_ws;
    float* qw = (float*)(ws);                                   // 64*512  f32
    float* V  = (float*)(ws + 131072);                          // 64*2056 f32
    float* cb = (float*)(ws + 131072 + 526336);                 // 64      f32
    float* S  = (float*)(ws + 131072 + 526336 + 256);           // 64*8    f32

    k_query_wmma<<<dim3(B_ / 16, DF_ / 16), 32, 0, stream>>>(text, Wt, bt, Wa, qw);
    k_possum<<<2, 256, 0, stream>>>(position, S);
    k_cb<<<1, 64, 0, stream>>>(qw, bo, ba, cb);
    k_vmat_wmma<<<dim3((DO_ + 15) / 16, B_ / 16), 32, 0, stream>>>(Wo, qw, V);
    k_final<<<B_ * (N_ / 8), 256, 0, stream>>>(visual, position, V, S, cb, out);
}